// DARFU_15857019257068
// MI455X (gfx1250) — compile-verified
//
#include <hip/hip_runtime.h>

typedef float v2f __attribute__((ext_vector_type(2)));
typedef float v8f __attribute__((ext_vector_type(8)));
typedef int   v4i __attribute__((vector_size(16)));

#define TAU_INV (1.0f / 1.5f)
#define EPS_DEN 1e-9f

typedef __attribute__((address_space(1))) v4i as1_v4i;
typedef __attribute__((address_space(3))) v4i as3_v4i;

#if defined(__HIP_DEVICE_COMPILE__) && \
    __has_builtin(__builtin_amdgcn_global_load_async_to_lds_b128) && \
    __has_builtin(__builtin_amdgcn_s_wait_asynccnt)
#define USE_ASYNC 1
#else
#define USE_ASYNC 0
#endif

__device__ __forceinline__ v8f wmma4(v2f a, v2f b, v8f c) {
  // D = A(16x4, f32) * B(4x16, f32) + C(16x16, f32)
  return __builtin_amdgcn_wmma_f32_16x16x4_f32(
      /*neg_a=*/false, a, /*neg_b=*/false, b,
      /*c_mod=*/(short)0, c, /*reuse_a=*/false, /*reuse_b=*/false);
}

__device__ __forceinline__ void wave_lds_fence() {
  __asm__ volatile("s_wait_dscnt 0" ::: "memory");
}

// 16-byte global -> LDS copy: async (ASYNCcnt) on gfx1250, sync fallback.
__device__ __forceinline__ void lds_copy16(void* lds, const void* g) {
#if USE_ASYNC
  __builtin_amdgcn_global_load_async_to_lds_b128(
      (as1_v4i*)g, (as3_v4i*)lds, 0, 0);
#else
  *reinterpret_cast<float4*>(lds) = *reinterpret_cast<const float4*>(g);
#endif
}

__device__ __forceinline__ void async_wait0() {
#if USE_ASYNC
  __builtin_amdgcn_s_wait_asynccnt(0);
#endif
  __asm__ volatile("" ::: "memory");
}

// ---------------------------------------------------------------------------
// Kernel A: affinity softmax + message aggregation.
// Block = 64 threads (2 waves). Each wave owns a 16-row tile (32 rows/block).
// Double-buffered 16-col panels streamed via async-to-LDS; per panel:
// Gram via f32 WMMA, E = exp, unnormalized m += E*S_panel, rowsum.
// Writes h = [S, m/rowsum] into H[BN, 512].
// ---------------------------------------------------------------------------
__global__ __launch_bounds__(64)
void k_affinity_aggregate(const float* __restrict__ S, float* __restrict__ H) {
  constexpr int N = 1024, D = 256, LDP = D + 4;
  __shared__ __align__(16) float sRows[32 * LDP];
  __shared__ __align__(16) float sPanel[2][16 * LDP];
  __shared__ float sE[2][16 * 17];
  __shared__ float sRowSq[32];
  __shared__ float sColSq[16];

  const int tid  = threadIdx.x;
  const int w    = tid >> 5;
  const int lane = tid & 31;
  const int aM   = lane & 15;
  const int hi   = lane >> 4;

  const int b    = blockIdx.y;
  const int row0 = blockIdx.x * 32;
  const float* Sb = S + (size_t)b * N * D;

  // Load 32 rows of S into LDS (sync: needed immediately).
  for (int i = tid; i < 32 * (D / 4); i += 64) {
    int r = i / (D / 4), c4 = i % (D / 4);
    float4 v = reinterpret_cast<const float4*>(Sb + (size_t)(row0 + r) * D)[c4];
    *reinterpret_cast<float4*>(&sRows[r * LDP + c4 * 4]) = v;
  }
  // Prefetch panel 0 (async path overlaps with rowSq / H-copy below).
#pragma unroll
  for (int j = 0; j < 16; ++j)
    lds_copy16(&sPanel[0][j * LDP + tid * 4], Sb + (size_t)j * D + tid * 4);
  __syncthreads();

  if (tid < 32) {
    float s = 0.f;
    const float* rp = &sRows[tid * LDP];
    for (int k = 0; k < D; ++k) s += rp[k] * rp[k];
    sRowSq[tid] = s;
  }
  // h[:, 0:256] = S (coalesced copy while rows are in LDS)
  for (int i = tid; i < 32 * (D / 4); i += 64) {
    int r = i / (D / 4), c4 = i % (D / 4);
    float4 v = *reinterpret_cast<const float4*>(&sRows[r * LDP + c4 * 4]);
    reinterpret_cast<float4*>(H + ((size_t)b * N + row0 + r) * (size_t)(2 * D))[c4] = v;
  }
  async_wait0();
  __syncthreads();  // panel 0 + rowSq ready

  v8f macc[16];
  const v8f vzero = {0.f, 0.f, 0.f, 0.f, 0.f, 0.f, 0.f, 0.f};
#pragma unroll
  for (int c = 0; c < 16; ++c) macc[c] = vzero;
  float rs[8] = {0.f, 0.f, 0.f, 0.f, 0.f, 0.f, 0.f, 0.f};

  const float* aBase = &sRows[(w * 16 + aM) * LDP];

  for (int p = 0; p < N / 16; ++p) {
    const int col0 = p * 16;
    const float* cur = sPanel[p & 1];

    if (tid < 16) {
      float s = 0.f;
      const float* cp = &cur[tid * LDP];
      for (int k = 0; k < D; ++k) s += cp[k] * cp[k];
      sColSq[tid] = s;
    }
    // Kick off next panel into the other buffer (overlaps with WMMA below).
    if (p + 1 < N / 16) {
      float* nxt = sPanel[(p + 1) & 1];
#pragma unroll
      for (int j = 0; j < 16; ++j)
        lds_copy16(&nxt[j * LDP + tid * 4],
                   Sb + (size_t)(col0 + 16 + j) * D + tid * 4);
    }
    __syncthreads();  // sColSq visible (cur buffer untouched by prefetch)

    // Gram tile: G = S_rows(16xD) x S_cols^T(Dx16)
    const float* nBase = &cur[aM * LDP];
    v8f g = vzero;
#pragma unroll 8
    for (int k0 = 0; k0 < D; k0 += 4) {
      v2f a  = {aBase[k0 + 2 * hi], aBase[k0 + 2 * hi + 1]};
      v2f bb = {nBase[k0 + 2 * hi], nBase[k0 + 2 * hi + 1]};
      g = wmma4(a, bb, g);
    }

    const float sqc = sColSq[aM];
#pragma unroll
    for (int v = 0; v < 8; ++v) {
      const int M = v + 8 * hi;
      float t = sRowSq[w * 16 + M] + sqc - 2.0f * g[v];
      float e = __expf(-t * TAU_INV);                      // arg <= 0: stable
      e = ((row0 + w * 16 + M) == (col0 + aM)) ? 0.f : e;  // off-diagonal mask
      sE[w][M * 17 + aM] = e;                              // A-operand layout
      rs[v] += e;
    }
    wave_lds_fence();

    // m += E(16x16) x S_cols(16xD)
    const float* eA = &sE[w][aM * 17];
#pragma unroll
    for (int c = 0; c < 16; ++c) {
      v8f acc = macc[c];
#pragma unroll
      for (int k0 = 0; k0 < 16; k0 += 4) {
        const int j = k0 + 2 * hi;
        v2f a  = {eA[j], eA[j + 1]};
        v2f bb = {cur[j * LDP + c * 16 + aM], cur[(j + 1) * LDP + c * 16 + aM]};
        acc = wmma4(a, bb, acc);
      }
      macc[c] = acc;
    }
    async_wait0();    // next panel landed
    __syncthreads();  // everyone done with cur
  }

  // Reduce rowsum over the 16 lanes of each half-wave, then invert.
#pragma unroll
  for (int v = 0; v < 8; ++v) {
    float s = rs[v];
    s += __shfl_xor(s, 1, 32);
    s += __shfl_xor(s, 2, 32);
    s += __shfl_xor(s, 4, 32);
    s += __shfl_xor(s, 8, 32);
    rs[v] = 1.0f / (s + EPS_DEN);
  }

  // Stage m tile through (now free) panel buffer, write coalesced b128s.
  float* stage = sPanel[w];
#pragma unroll
  for (int c = 0; c < 16; ++c) {
#pragma unroll
    for (int v = 0; v < 8; ++v)
      stage[(v + 8 * hi) * LDP + c * 16 + aM] = macc[c][v] * rs[v];
  }
  wave_lds_fence();
  float* Hm = H + ((size_t)b * N + row0 + w * 16) * (size_t)(2 * D) + D;
#pragma unroll
  for (int j = 0; j < 32; ++j) {  // 16 rows x 64 float4 / 32 lanes
    int i = lane + 32 * j;
    int r = i / (D / 4), c4 = i % (D / 4);
    float4 v = *reinterpret_cast<const float4*>(&stage[r * LDP + c4 * 4]);
    reinterpret_cast<float4*>(Hm + (size_t)r * (2 * D))[c4] = v;
  }
}

// ---------------------------------------------------------------------------
// Kernel B: S_next = S_cur + relu(h @ W1 + b1) @ W2 + b2
// Block = 32 threads (1 wave) owning a 16-row tile of h=[BN, 512].
// ---------------------------------------------------------------------------
__global__ __launch_bounds__(32)
void k_mlp_update(const float* __restrict__ H,
                  const float* __restrict__ W1, const float* __restrict__ b1,
                  const float* __restrict__ W2, const float* __restrict__ b2,
                  const float* __restrict__ Scur, float* __restrict__ Snext) {
  constexpr int D = 256, DH = 512, LDX = DH + 4, LDH = D + 4;
  __shared__ __align__(16) float sX[16 * LDX];
  __shared__ float sHid[16 * LDH];

  const int lane = threadIdx.x & 31;
  const int aM   = lane & 15;
  const int hi   = lane >> 4;
  const size_t gr0 = (size_t)blockIdx.x * 16;

#pragma unroll
  for (int j = 0; j < 64; ++j) {  // 16 rows x 128 float4 / 32 lanes
    int i = lane + 32 * j;
    int r = i / (DH / 4), c4 = i % (DH / 4);
    lds_copy16(&sX[r * LDX + c4 * 4], H + (gr0 + r) * DH + c4 * 4);
  }
  async_wait0();
  __syncthreads();

  const float* aX = &sX[aM * LDX];
  // hid = relu(X @ W1 + b1), K = 512
#pragma unroll 1
  for (int c = 0; c < 16; ++c) {
    const float bb = b1[c * 16 + aM];
    v8f acc = {bb, bb, bb, bb, bb, bb, bb, bb};
#pragma unroll 4
    for (int k0 = 0; k0 < DH; k0 += 4) {
      const int k = k0 + 2 * hi;
      v2f a  = {aX[k], aX[k + 1]};
      v2f bw = {W1[(size_t)k * D + c * 16 + aM], W1[(size_t)(k + 1) * D + c * 16 + aM]};
      acc = wmma4(a, bw, acc);
    }
#pragma unroll
    for (int v = 0; v < 8; ++v) {
      float hv = acc[v] > 0.f ? acc[v] : 0.f;
      sHid[(v + 8 * hi) * LDH + c * 16 + aM] = hv;
    }
  }
  wave_lds_fence();

  const float* aH = &sHid[aM * LDH];
  float* stage = sX;  // sX is dead after GEMM-1; reuse for output staging
  // out = hid @ W2 + b2  (staged to LDS, then coalesced residual-add store)
#pragma unroll 1
  for (int c = 0; c < 16; ++c) {
    const float bb = b2[c * 16 + aM];
    v8f acc = {bb, bb, bb, bb, bb, bb, bb, bb};
#pragma unroll 4
    for (int k0 = 0; k0 < D; k0 += 4) {
      const int k = k0 + 2 * hi;
      v2f a  = {aH[k], aH[k + 1]};
      v2f bw = {W2[(size_t)k * D + c * 16 + aM], W2[(size_t)(k + 1) * D + c * 16 + aM]};
      acc = wmma4(a, bw, acc);
    }
#pragma unroll
    for (int v = 0; v < 8; ++v)
      stage[(v + 8 * hi) * LDX + c * 16 + aM] = acc[v];
  }
  wave_lds_fence();

#pragma unroll
  for (int j = 0; j < 32; ++j) {  // 16 rows x 64 float4 / 32 lanes
    int i = lane + 32 * j;
    int r = i / (D / 4), c4 = i % (D / 4);
    float4 u = *reinterpret_cast<const float4*>(&stage[r * LDX + c4 * 4]);
    float4 s = reinterpret_cast<const float4*>(Scur + (gr0 + r) * D)[c4];
    float4 o = {s.x + u.x, s.y + u.y, s.z + u.z, s.w + u.w};
    reinterpret_cast<float4*>(Snext + (gr0 + r) * D)[c4] = o;
  }
}

// ---------------------------------------------------------------------------
// Kernel C: T_final[b,i,j] = ||s_i||^2 + ||s_j||^2 - 2 s_i.s_j
// Same streaming structure as kernel A; T tiles staged in LDS for b128 stores.
// ---------------------------------------------------------------------------
__global__ __launch_bounds__(64)
void k_pairwise_T(const float* __restrict__ S, float* __restrict__ T) {
  constexpr int N = 1024, D = 256, LDP = D + 4, LDT = 20;
  __shared__ __align__(16) float sRows[32 * LDP];
  __shared__ __align__(16) float sPanel[2][16 * LDP];
  __shared__ __align__(16) float sT[2][16 * LDT];
  __shared__ float sRowSq[32];
  __shared__ float sColSq[16];

  const int tid  = threadIdx.x;
  const int w    = tid >> 5;
  const int lane = tid & 31;
  const int aM   = lane & 15;
  const int hi   = lane >> 4;

  const int b    = blockIdx.y;
  const int row0 = blockIdx.x * 32;
  const float* Sb = S + (size_t)b * N * D;
  float* Tb = T + (size_t)b * N * N;

  for (int i = tid; i < 32 * (D / 4); i += 64) {
    int r = i / (D / 4), c4 = i % (D / 4);
    float4 v = reinterpret_cast<const float4*>(Sb + (size_t)(row0 + r) * D)[c4];
    *reinterpret_cast<float4*>(&sRows[r * LDP + c4 * 4]) = v;
  }
#pragma unroll
  for (int j = 0; j < 16; ++j)
    lds_copy16(&sPanel[0][j * LDP + tid * 4], Sb + (size_t)j * D + tid * 4);
  __syncthreads();

  if (tid < 32) {
    float s = 0.f;
    const float* rp = &sRows[tid * LDP];
    for (int k = 0; k < D; ++k) s += rp[k] * rp[k];
    sRowSq[tid] = s;
  }
  async_wait0();
  __syncthreads();

  const float* aBase = &sRows[(w * 16 + aM) * LDP];
  const v8f vzero = {0.f, 0.f, 0.f, 0.f, 0.f, 0.f, 0.f, 0.f};

  for (int p = 0; p < N / 16; ++p) {
    const int col0 = p * 16;
    const float* cur = sPanel[p & 1];

    if (tid < 16) {
      float s = 0.f;
      const float* cp = &cur[tid * LDP];
      for (int k = 0; k < D; ++k) s += cp[k] * cp[k];
      sColSq[tid] = s;
    }
    if (p + 1 < N / 16) {
      float* nxt = sPanel[(p + 1) & 1];
#pragma unroll
      for (int j = 0; j < 16; ++j)
        lds_copy16(&nxt[j * LDP + tid * 4],
                   Sb + (size_t)(col0 + 16 + j) * D + tid * 4);
    }
    __syncthreads();

    const float* nBase = &cur[aM * LDP];
    v8f g = vzero;
#pragma unroll 8
    for (int k0 = 0; k0 < D; k0 += 4) {
      v2f a  = {aBase[k0 + 2 * hi], aBase[k0 + 2 * hi + 1]};
      v2f bb = {nBase[k0 + 2 * hi], nBase[k0 + 2 * hi + 1]};
      g = wmma4(a, bb, g);
    }
    const float sqc = sColSq[aM];
#pragma unroll
    for (int v = 0; v < 8; ++v) {
      const int M = v + 8 * hi;
      sT[w][M * LDT + aM] = sRowSq[w * 16 + M] + sqc - 2.0f * g[v];
    }
    wave_lds_fence();
    // Coalesced tile store: 16 rows x 4 float4 per wave, 2 per lane.
#pragma unroll
    for (int jj = 0; jj < 2; ++jj) {
      int i = lane + 32 * jj;
      int r = i >> 2, q = i & 3;
      float4 v = *reinterpret_cast<const float4*>(&sT[w][r * LDT + q * 4]);
      *reinterpret_cast<float4*>(&Tb[(size_t)(row0 + w * 16 + r) * N + col0 + q * 4]) = v;
    }
    async_wait0();
    __syncthreads();
  }
}

// ---------------------------------------------------------------------------
extern "C" void kernel_launch(void* const* d_in, const int* in_sizes, int n_in,
                              void* d_out, int out_size, void* d_ws, size_t ws_size,
                              hipStream_t stream) {
  (void)in_sizes; (void)n_in; (void)out_size; (void)d_ws; (void)ws_size;
  const float* S0 = (const float*)d_in[0];
  const float* W1 = (const float*)d_in[1];
  const float* b1 = (const float*)d_in[2];
  const float* W2 = (const float*)d_in[3];
  const float* b2 = (const float*)d_in[4];
  float* out = (float*)d_out;

  constexpr size_t B = 32, N = 1024, D = 256;
  float* S_out = out;                         // [B,N,D]   final S_M
  float* T_out = out + B * N * D;             // [B,N,N]   final T
  // Scratch carved out of the (not-yet-needed) T region: H (64 MB) + S ping (32 MB).
  float* Hbuf  = T_out;                       // [B*N, 2D]
  float* S_alt = T_out + B * N * 2 * D;       // [B,N,D]

  const float* Scur = S0;
  dim3 gridA(N / 32, B);
  for (int it = 0; it < 4; ++it) {
    k_affinity_aggregate<<<gridA, 64, 0, stream>>>(Scur, Hbuf);
    float* Snext = (it & 1) ? S_out : S_alt;  // it=3 lands in S_out
    k_mlp_update<<<(int)((B * N) / 16), 32, 0, stream>>>(Hbuf, W1, b1, W2, b2, Scur, Snext);
    Scur = Snext;
  }
  k_pairwise_T<<<gridA, 64, 0, stream>>>(S_out, T_out);
}